// GAEChamferLoss_2714419331076
// MI455X (gfx1250) — compile-verified
//
#include <hip/hip_runtime.h>
#include <hip/hip_bf16.h>

typedef __attribute__((ext_vector_type(2))) float v2f;
typedef __attribute__((ext_vector_type(8))) float v8f;

#define BIGV 1.0e10f

// Row-min kernel: for direction dir (blockIdx.y):
//   dir==0: rows = x points, cols = y points  -> minsA ("mins2" in reference)
//   dir==1: rows = y points, cols = x points  -> minsB ("mins1" in reference)
// Each wave (32 lanes) owns a 16-row tile; 8 waves per 256-thread block -> 128 rows/block.
// Streams all N columns in 16-wide tiles, one V_WMMA_F32_16X16X4_F32 per tile.
__global__ __launch_bounds__(256)
void chamfer_rowmin_kernel(const float* __restrict__ x,
                           const float* __restrict__ y,
                           const int* __restrict__ mask,
                           float* __restrict__ minsA,
                           float* __restrict__ minsB,
                           int N)
{
    const int b   = blockIdx.z;
    const int dir = blockIdx.y;

    const float* Arows = (dir == 0) ? x : y;   // row point cloud (B,3,N)
    const float* Bcols = (dir == 0) ? y : x;   // column point cloud
    float* out         = (dir == 0) ? minsA : minsB;

    const float* A0 = Arows + (size_t)(b * 3 + 0) * N;
    const float* A1 = Arows + (size_t)(b * 3 + 1) * N;
    const float* A2 = Arows + (size_t)(b * 3 + 2) * N;
    const float* B0 = Bcols + (size_t)(b * 3 + 0) * N;
    const float* B1 = Bcols + (size_t)(b * 3 + 1) * N;
    const float* B2 = Bcols + (size_t)(b * 3 + 2) * N;
    const int*   MK = mask + (size_t)b * N;

    const int tid  = threadIdx.x;
    const int wave = tid >> 5;      // 0..7
    const int lane = tid & 31;      // wave32
    const int l16  = lane & 15;     // column / row-in-tile index
    const int hi   = lane >> 4;     // 0: K=0,1 half; 1: K=2,3 half

    const int row_base = blockIdx.x * 128 + wave * 16;

    // ---- A fragment: 16x4 f32, K padded 3->4 with zero ----
    const int p = row_base + l16;
    const float a0 = A0[p];
    const float a1 = A1[p];
    const float a2 = A2[p];
    v2f afrag;
    afrag.x = hi ? a2   : a0;
    afrag.y = hi ? 0.0f : a1;

    // ---- per-lane row norms for the 8 rows this lane accumulates:
    //      D VGPR v, this lane -> row (row_base + v + 8*hi)
    float xxr[8];
#pragma unroll
    for (int v = 0; v < 8; ++v) {
        const int r = row_base + 8 * hi + v;
        const float r0 = A0[r], r1 = A1[r], r2 = A2[r];
        xxr[v] = r0 * r0 + r1 * r1 + r2 * r2;
    }

    float acc[8];
#pragma unroll
    for (int v = 0; v < 8; ++v) acc[v] = 3.0e38f;

    // ---- stream column tiles ----
    for (int mb = 0; mb < N; mb += 16) {
        const int q = mb + l16;
        const float b0v = B0[q];
        const float b1v = B1[q];
        const float b2v = B2[q];

        // prefetch next tile's coordinate lines (global_prefetch_b8)
        __builtin_prefetch(B0 + q + 16, 0, 0);
        __builtin_prefetch(B1 + q + 16, 0, 0);
        __builtin_prefetch(B2 + q + 16, 0, 0);

        // masked column norm: invalid columns pushed to ~BIG
        float yyq = b0v * b0v + b1v * b1v + b2v * b2v;
        yyq += MK[q] ? 0.0f : BIGV;

        // B fragment: 4x16 f32, K padded
        v2f bfrag;
        bfrag.x = hi ? b2v  : b0v;
        bfrag.y = hi ? 0.0f : b1v;

        v8f c = {0.f, 0.f, 0.f, 0.f, 0.f, 0.f, 0.f, 0.f};
        // D = A(16x4) x B(4x16) + 0 : Gram tile of inner products
        v8f d = __builtin_amdgcn_wmma_f32_16x16x4_f32(
            /*neg_a=*/false, afrag, /*neg_b=*/false, bfrag,
            /*c_mod=*/(short)0, c, /*reuse_a=*/false, /*reuse_b=*/false);

#pragma unroll
        for (int v = 0; v < 8; ++v) {
            const float P = xxr[v] + yyq - 2.0f * d[v];
            acc[v] = fminf(acc[v], P);
        }
    }

    // ---- reduce across the 16 columns held by lanes (per wave) via LDS ----
    __shared__ float red[8][16][17];   // [wave][row-in-tile][col], padded
#pragma unroll
    for (int v = 0; v < 8; ++v) {
        red[wave][v + 8 * hi][l16] = acc[v];
    }
    __syncthreads();

    if (lane < 16) {
        float mn = red[wave][lane][0];
#pragma unroll
        for (int i = 1; i < 16; ++i) mn = fminf(mn, red[wave][lane][i]);
        out[(size_t)b * N + row_base + lane] = mn;
    }
}

// Final reduction: masked means per batch for both directions, then
// mean over batches of (means1 + means2). One block, 16 threads per batch.
__global__ __launch_bounds__(256)
void chamfer_reduce_kernel(const float* __restrict__ minsA,
                           const float* __restrict__ minsB,
                           const int* __restrict__ mask,
                           float* __restrict__ out,
                           int N)
{
    const int t    = threadIdx.x;     // 256 threads: 16 batches x 16 lanes
    const int b    = t >> 4;
    const int l16  = t & 15;

    float sA = 0.0f, sB = 0.0f, cnt = 0.0f;
    for (int n = l16; n < N; n += 16) {
        const size_t idx = (size_t)b * N + n;
        const float m = (float)mask[idx];
        sA  += minsA[idx] * m;
        sB  += minsB[idx] * m;
        cnt += m;
    }

    __shared__ float shA[256], shB[256], shC[256];
    shA[t] = sA; shB[t] = sB; shC[t] = cnt;
    __syncthreads();

    __shared__ float perBatch[16];
    if (l16 == 0) {
        float tA = 0.0f, tB = 0.0f, tC = 0.0f;
#pragma unroll
        for (int i = 0; i < 16; ++i) {
            tA += shA[t + i];
            tB += shB[t + i];
            tC += shC[t + i];
        }
        perBatch[b] = tB / tC + tA / tC;   // means1[b] + means2[b]
    }
    __syncthreads();

    if (t == 0) {
        float r = 0.0f;
#pragma unroll
        for (int i = 0; i < 16; ++i) r += perBatch[i];
        out[0] = r / 16.0f;                // mean over batches
    }
}

extern "C" void kernel_launch(void* const* d_in, const int* in_sizes, int n_in,
                              void* d_out, int out_size, void* d_ws, size_t ws_size,
                              hipStream_t stream)
{
    const float* x    = (const float*)d_in[0];   // (B,3,N) f32
    const float* y    = (const float*)d_in[1];   // (B,3,N) f32
    const int*   mask = (const int*)d_in[2];     // (B,N) i32

    const int Bb = 16;
    const int BN = in_sizes[2];      // B*N
    const int N  = BN / Bb;          // 4096

    float* minsA = (float*)d_ws;     // (B,N): dir 0 (rows = x)
    float* minsB = minsA + BN;       // (B,N): dir 1 (rows = y)

    dim3 grid(N / 128, 2, Bb);       // 32 row-blocks x 2 directions x 16 batches
    chamfer_rowmin_kernel<<<grid, 256, 0, stream>>>(x, y, mask, minsA, minsB, N);
    chamfer_reduce_kernel<<<1, 256, 0, stream>>>(minsA, minsB, mask, (float*)d_out, N);
}